// GaussianProcessRiemmanianMetric_54520314856270
// MI455X (gfx1250) — compile-verified
//
#include <hip/hip_runtime.h>

typedef float v2f __attribute__((ext_vector_type(2)));
typedef float v8f __attribute__((ext_vector_type(8)));

#define WAVES_PER_BLOCK 4
#define PTS_PER_WAVE    32
#define PTS_PER_BLOCK   (WAVES_PER_BLOCK * PTS_PER_WAVE)
#define NB              64
#define LDS_STRIDE      68   // floats; row = 272B (16B aligned), conflict-free stores

// Christoffel symbols of GP-parametrized Riemannian metric.
// Stage 1: proj = base_point @ Omega + phase  via V_WMMA_F32_16X16X4_F32 (K=3 padded to 4).
// Stage 2: per-lane (one point per lane) GP eval/deriv + 3x3 Cholesky/Koszul algebra.
__global__ __launch_bounds__(PTS_PER_BLOCK)
void christoffel_gp_kernel(const float* __restrict__ bp,   // [N,3]
                           const float* __restrict__ Om,   // [3,64]
                           const float* __restrict__ ph,   // [64]
                           const float* __restrict__ be,   // [64,6]
                           float* __restrict__ out,        // [N,3,3,3]
                           int N)
{
    __shared__ float smem[WAVES_PER_BLOCK][PTS_PER_WAVE][LDS_STRIDE];

    const int tid  = threadIdx.x;
    const int w    = tid >> 5;       // wave in block
    const int lane = tid & 31;
    const int hi   = lane >> 4;      // 0: lanes 0-15, 1: lanes 16-31
    const int c    = lane & 15;      // column within 16-wide tile
    const int waveBase = blockIdx.x * PTS_PER_BLOCK + w * PTS_PER_WAVE;

    // ---------------- Stage 1: WMMA projection ----------------
    // B operand (Omega, 4x16 tiles; K rows: VGPR0={K0 lo,K2 hi}, VGPR1={K1 lo,K3 hi})
    v2f  bv[4];
    float cph[4];
#pragma unroll
    for (int t = 0; t < 4; ++t) {
        int cg = 16 * t + c;
        float o0 = Om[cg], o1 = Om[NB + cg], o2 = Om[2 * NB + cg];
        bv[t][0] = hi ? o2   : o0;
        bv[t][1] = hi ? 0.0f : o1;
        cph[t]   = ph[cg];
    }

#pragma unroll
    for (int mt = 0; mt < 2; ++mt) {
        int pt  = waveBase + 16 * mt + c;
        int ptc = (pt < N) ? pt : (N - 1);        // clamp (no EXEC divergence)
        const float* row = bp + (size_t)ptc * 3;
        float a0 = row[0], a1 = row[1], a2 = row[2];
        // A operand (16x4): lanes 0-15 hold K0,K1 = x0,x1; lanes 16-31 hold K2,K3 = x2,0
        v2f av;
        av[0] = hi ? a2   : a0;
        av[1] = hi ? 0.0f : a1;
#pragma unroll
        for (int t = 0; t < 4; ++t) {
            v8f cv = {cph[t], cph[t], cph[t], cph[t], cph[t], cph[t], cph[t], cph[t]};
            v8f d  = __builtin_amdgcn_wmma_f32_16x16x4_f32(
                         /*neg_a=*/false, av, /*neg_b=*/false, bv[t],
                         /*c_mod=*/(short)0, cv, /*reuse_a=*/false, /*reuse_b=*/false);
            // D layout: VGPR r, lane l -> row (r + 8*hi), col (l&15) of the 16x16 tile
            float* sb = &smem[w][16 * mt + 8 * hi][16 * t + c];
#pragma unroll
            for (int r = 0; r < 8; ++r) sb[r * LDS_STRIDE] = d[r];
        }
    }
    __syncthreads();

    // ---------------- Stage 2: per-lane GP accumulation ----------------
    float ge[6]    = {0.f, 0.f, 0.f, 0.f, 0.f, 0.f};
    float gd[6][3] = {};

    const float4* rowp = reinterpret_cast<const float4*>(&smem[w][lane][0]);
#pragma unroll 2
    for (int mm = 0; mm < NB / 4; ++mm) {
        float4 pv = rowp[mm];
        float pa[4] = {pv.x, pv.y, pv.z, pv.w};
#pragma unroll
        for (int q = 0; q < 4; ++q) {
            int   m  = mm * 4 + q;                  // uniform -> s_load operands
            float p  = pa[q];
            float s  = __sinf(p);
            float co = __cosf(p);
            float cw0 = co * Om[m];
            float cw1 = co * Om[NB + m];
            float cw2 = co * Om[2 * NB + m];
#pragma unroll
            for (int g = 0; g < 6; ++g) {
                float bg = be[m * 6 + g];
                ge[g]    = fmaf(s,   bg, ge[g]);
                gd[g][0] = fmaf(cw0, bg, gd[g][0]);
                gd[g][1] = fmaf(cw1, bg, gd[g][1]);
                gd[g][2] = fmaf(cw2, bg, gd[g][2]);
            }
        }
    }

    // ---------------- Stage 3: 3x3 Christoffel algebra ----------------
    // tril order g: 0->(0,0) 1->(1,0) 2->(1,1) 3->(2,0) 4->(2,1) 5->(2,2)
    float d0 = __expf(0.5f * ge[0]);
    float d1 = __expf(0.5f * ge[2]);
    float d2 = __expf(0.5f * ge[5]);

    float Lm[3][3] = {{d0, 0.f, 0.f}, {ge[1], d1, 0.f}, {ge[3], ge[4], d2}};

    // dL with the reference's jnp.diagonal(axis1=1, axis2=3) semantics:
    //   dL[a][a][t] = 0.5 * d_a * dL_old[t][a][t]
    float Dm[3][3][3] = {};
#pragma unroll
    for (int k = 0; k < 3; ++k) {
        Dm[1][0][k] = gd[1][k];
        Dm[2][0][k] = gd[3][k];
        Dm[2][1][k] = gd[4][k];
    }
    Dm[0][0][0] = 0.5f * d0 * gd[0][0];
    Dm[0][0][1] = 0.5f * d0 * gd[1][1];
    Dm[0][0][2] = 0.5f * d0 * gd[3][2];
    Dm[1][1][0] = 0.f;
    Dm[1][1][1] = 0.5f * d1 * gd[2][1];
    Dm[1][1][2] = 0.5f * d1 * gd[4][2];
    Dm[2][2][0] = 0.f;
    Dm[2][2][1] = 0.f;
    Dm[2][2][2] = 0.5f * d2 * gd[5][2];

    // partials[i][l][k] = sum_j L[i][j]*dL[l][j][k] + dL[i][j][k]*L[l][j]
    float P[3][3][3];
#pragma unroll
    for (int i = 0; i < 3; ++i)
#pragma unroll
        for (int l = 0; l < 3; ++l)
#pragma unroll
            for (int k = 0; k < 3; ++k) {
                float acc = 0.f;
#pragma unroll
                for (int j = 0; j < 3; ++j)
                    acc += Lm[i][j] * Dm[l][j][k] + Dm[i][j][k] * Lm[l][j];
                P[i][l][k] = acc;
            }

    // cometric G = L^{-T} L^{-1}
    float i00 = 1.0f / d0, i11 = 1.0f / d1, i22 = 1.0f / d2;
    float i10 = -Lm[1][0] * i00 * i11;
    float i21 = -Lm[2][1] * i11 * i22;
    float i20 = -(Lm[2][0] * i00 + Lm[2][1] * i10) * i22;
    float G[3][3];
    G[0][0] = i00 * i00 + i10 * i10 + i20 * i20;
    G[0][1] = i10 * i11 + i20 * i21;  G[1][0] = G[0][1];
    G[0][2] = i20 * i22;              G[2][0] = G[0][2];
    G[1][1] = i11 * i11 + i21 * i21;
    G[1][2] = i21 * i22;              G[2][1] = G[1][2];
    G[2][2] = i22 * i22;

    // Gamma^k_{ij} = 0.5 * sum_l G[l][k] * (P[j][l][i] + P[l][i][j] - P[i][j][l])
    int ptg = waveBase + lane;
    if (ptg < N) {
        float* o = out + (size_t)ptg * 27;
#pragma unroll
        for (int k = 0; k < 3; ++k)
#pragma unroll
            for (int i = 0; i < 3; ++i)
#pragma unroll
                for (int j = 0; j < 3; ++j) {
                    float acc = 0.f;
#pragma unroll
                    for (int l = 0; l < 3; ++l)
                        acc += G[l][k] * (P[j][l][i] + P[l][i][j] - P[i][j][l]);
                    o[(k * 3 + i) * 3 + j] = 0.5f * acc;
                }
    }
}

extern "C" void kernel_launch(void* const* d_in, const int* in_sizes, int n_in,
                              void* d_out, int out_size, void* d_ws, size_t ws_size,
                              hipStream_t stream) {
    const float* bp = (const float*)d_in[0];  // base_point [N,3]
    const float* Om = (const float*)d_in[1];  // Omega [3,64]
    const float* ph = (const float*)d_in[2];  // phase [64]
    const float* be = (const float*)d_in[3];  // beta [64,6]
    float* out = (float*)d_out;

    int N = in_sizes[0] / 3;
    int blocks = (N + PTS_PER_BLOCK - 1) / PTS_PER_BLOCK;
    hipLaunchKernelGGL(christoffel_gp_kernel, dim3(blocks), dim3(PTS_PER_BLOCK), 0, stream,
                       bp, Om, ph, be, out, N);
}